// SA_Layer_42167988912479
// MI455X (gfx1250) — compile-verified
//
#include <hip/hip_runtime.h>
#include <math.h>

#define B_  4
#define C_  256
#define N_  512
#define CQ_ 64

typedef __attribute__((ext_vector_type(2))) float v2f;
typedef __attribute__((ext_vector_type(8))) float v8f;

// ---------------------------------------------------------------------------
// LDS-staged WMMA GEMM:  C = alpha * A(MxK) @ B(KxN)  [+ Cold] [+ bias(m)]
// Generic element strides; per-batch offsets abatch/bbatch/cbatch.
// Block = 128 threads = 4 waves; block tile = 64x64; wave tile = 16x64
// (4 v8f accumulators, A fragment reused across 4 WMMAs per K-step).
// K is consumed in 16-wide LDS panels; next panel prefetched to cache
// via global_prefetch. Requires M%64==0, N%64==0, K%16==0.
// Fragment layouts follow CDNA5 ISA 7.12.2 (16x4 A, 4x16 B, 16x16 C/D).
// ---------------------------------------------------------------------------
#define KB_  16
#define LDP_ 68   // padded LDS row stride (floats) to avoid bank conflicts

__global__ __launch_bounds__(128) void gemm64_wmma(
    const float* __restrict__ A, long ars, long acs, long abatch,
    const float* __restrict__ Bm, long brs, long bcs, long bbatch,
    float* __restrict__ Cm, long crs, long ccs, long cbatch,
    int M, int Nn, int K, float alpha, int accumulate,
    const float* __restrict__ bias)
{
    __shared__ float sA[KB_][LDP_];   // sA[k][m]  (A panel, K-major)
    __shared__ float sB[KB_][LDP_];   // sB[k][n]  (B panel, K-major)

    const int tid  = threadIdx.x;
    const int wave = tid >> 5;
    const int lane = tid & 31;
    const int half = lane >> 4;       // K-pair / row-half selector
    const int l16  = lane & 15;
    const int mblk = blockIdx.y * 64;
    const int nblk = blockIdx.x * 64;
    const int b    = blockIdx.z;

    const float* Ab = A  + (long)b * abatch;
    const float* Bb = Bm + (long)b * bbatch;
    float*       Cb = Cm + (long)b * cbatch;

    v8f acc[4] = {};                  // rows [mblk+wave*16, +16) x cols [nblk, +64)
    const int mrow = wave * 16 + l16;

    for (int kp = 0; kp < K; kp += KB_) {
        // Cooperative panel load: 1024 elements each, 8 per thread.
        #pragma unroll
        for (int e = 0; e < 8; ++e) {
            const int idx = tid + e * 128;   // 0..1023
            const int kk  = idx >> 6;        // 0..15
            const int mm  = idx & 63;        // 0..63
            sA[kk][mm] = Ab[(long)(mblk + mm) * ars + (long)(kp + kk) * acs];
            sB[kk][mm] = Bb[(long)(kp + kk) * brs + (long)(nblk + mm) * bcs];
        }
        __syncthreads();

        // Warm caches for the next K panel (gfx1250 global_prefetch path).
        if (kp + KB_ < K) {
            __builtin_prefetch(&Ab[(long)(mblk + (tid & 63)) * ars +
                                   (long)(kp + KB_ + (tid >> 6) * 8) * acs], 0, 1);
            __builtin_prefetch(&Bb[(long)(kp + KB_ + (tid >> 6) * 8) * brs +
                                   (long)(nblk + (tid & 63)) * bcs], 0, 1);
        }

        #pragma unroll
        for (int k = 0; k < KB_; k += 4) {
            const int ka = k + 2 * half;
            v2f af;                              // A 16x4 fragment: row l16, K=ka,ka+1
            af.x = sA[ka + 0][mrow];
            af.y = sA[ka + 1][mrow];
            #pragma unroll
            for (int ns = 0; ns < 4; ++ns) {     // reuse A across 4 column sub-tiles
                v2f bf;                          // B 4x16 fragment: col l16, K=ka,ka+1
                bf.x = sB[ka + 0][ns * 16 + l16];
                bf.y = sB[ka + 1][ns * 16 + l16];
                acc[ns] = __builtin_amdgcn_wmma_f32_16x16x4_f32(
                              false, af, false, bf, (short)0, acc[ns], false, false);
            }
        }
        __syncthreads();
    }

    // D layout: vgpr r holds row (r + 8*half), col l16 (within each 16x16 sub-tile)
    const int m0 = mblk + wave * 16;
    #pragma unroll
    for (int ns = 0; ns < 4; ++ns) {
        #pragma unroll
        for (int r = 0; r < 8; ++r) {
            const int m = m0 + r + 8 * half;
            const long idx = (long)m * crs + (long)(nblk + ns * 16 + l16) * ccs;
            float v = alpha * acc[ns][r];
            if (accumulate) v += Cb[idx];
            if (bias)       v += bias[m];
            Cb[idx] = v;
        }
    }
}

// ---------------------------------------------------------------------------
// e2 fold: Ksum[b][c][m] += (sincos-embed(angle_pca[b,m]) @ emb2_W^T + emb2_b)[c]
// One 64-thread block per (b,m); lanes 0..31 build the interleaved embedding.
// ---------------------------------------------------------------------------
__global__ __launch_bounds__(64) void e2_accum_kernel(
    const float* __restrict__ angle_pca, const float* __restrict__ W2,
    const float* __restrict__ b2, float* __restrict__ ksum)
{
    const int bm = blockIdx.x;
    const int b = bm >> 9, m = bm & (N_ - 1);
    const int t = threadIdx.x;
    __shared__ float emb[CQ_];
    const float ang = angle_pca[bm] * 3.8197186342054885f;  // 180/(pi*sigma)
    if (t < 32) {
        const float w = ang * __expf(-0.28782313662425575f * (float)t); // div_term
        emb[2 * t]     = __sinf(w);
        emb[2 * t + 1] = __cosf(w);
    }
    __syncthreads();
    float v = b2[t];
    #pragma unroll 8
    for (int d = 0; d < CQ_; ++d) v += emb[d] * W2[t * CQ_ + d];
    ksum[(long)b * CQ_ * N_ + (long)t * N_ + m] += v;
}

// qb[b,n] = x_q[b,n,:] . emb1_b
__global__ void qb_kernel(const float* __restrict__ xq,
                          const float* __restrict__ e1b, float* __restrict__ qb)
{
    const int i = blockIdx.x * blockDim.x + threadIdx.x;
    if (i >= B_ * N_) return;
    float s = 0.f;
    #pragma unroll 8
    for (int c = 0; c < CQ_; ++c) s += xq[(long)i * CQ_ + c] * e1b[c];
    qb[i] = s;
}

// ---------------------------------------------------------------------------
// Folded angle_lrf term: energy[b,n,m] += qb[b,n] + sum_i sin(w_i)q1[2i]+cos(w_i)q1[2i+1]
// One 256-thread block per (b,n); q1 row + div_term staged in LDS.
// ---------------------------------------------------------------------------
__global__ __launch_bounds__(256) void angle_add_kernel(
    const float* __restrict__ angle_lrf, const float* __restrict__ q1,
    const float* __restrict__ qb, float* __restrict__ energy)
{
    const int bn = blockIdx.x;               // b*N + n
    const int t = threadIdx.x;
    __shared__ float sq1[CQ_];
    __shared__ float sdiv[32];
    __shared__ float sqb;
    if (t < CQ_) sq1[t] = q1[(long)bn * CQ_ + t];
    if (t < 32)  sdiv[t] = __expf(-0.28782313662425575f * (float)t);
    if (t == 0)  sqb = qb[bn];
    __syncthreads();
    const float* arow = angle_lrf + (long)bn * N_;
    float*       erow = energy    + (long)bn * N_;
    const float qbv = sqb;
    #pragma unroll
    for (int j = 0; j < 2; ++j) {
        const int m = t + j * 256;
        const float base = arow[m] * 3.8197186342054885f;
        float acc = qbv;
        #pragma unroll 8
        for (int i = 0; i < 32; ++i) {
            const float w = base * sdiv[i];
            acc += __sinf(w) * sq1[2 * i] + __cosf(w) * sq1[2 * i + 1];
        }
        erow[m] += acc;
    }
}

// Row softmax over last axis (row length 512, 256 threads x 2 elems).
__global__ __launch_bounds__(256) void softmax_kernel(float* __restrict__ energy)
{
    float* row = energy + (long)blockIdx.x * N_;
    const int t = threadIdx.x;
    __shared__ float red[256];
    const float v0 = row[t], v1 = row[t + 256];
    red[t] = fmaxf(v0, v1);
    __syncthreads();
    for (int s = 128; s > 0; s >>= 1) {
        if (t < s) red[t] = fmaxf(red[t], red[t + s]);
        __syncthreads();
    }
    const float mx = red[0];
    __syncthreads();
    const float e0 = __expf(v0 - mx), e1 = __expf(v1 - mx);
    red[t] = e0 + e1;
    __syncthreads();
    for (int s = 128; s > 0; s >>= 1) {
        if (t < s) red[t] += red[t + s];
        __syncthreads();
    }
    const float inv = 1.f / red[0];
    row[t] = e0 * inv;
    row[t + 256] = e1 * inv;
}

// colsum[b,m] = sum_n energy[b,n,m]  (coalesced: consecutive threads -> consecutive m)
__global__ void colsum_kernel(const float* __restrict__ energy, float* __restrict__ colsum)
{
    const int i = blockIdx.x * blockDim.x + threadIdx.x;
    if (i >= B_ * N_) return;
    const int b = i >> 9, m = i & (N_ - 1);
    const float* base = energy + (long)b * N_ * N_ + m;
    float s = 0.f;
    for (int n = 0; n < N_; ++n) s += base[(long)n * N_];
    colsum[i] = s;
}

__global__ void colnorm_kernel(float* __restrict__ energy, const float* __restrict__ colsum)
{
    const long idx = (long)blockIdx.x * blockDim.x + threadIdx.x;
    if (idx >= (long)B_ * N_ * N_) return;
    const int b = (int)(idx / ((long)N_ * N_));
    const int m = (int)(idx & (N_ - 1));
    energy[idx] *= 1.f / (1e-12f + colsum[b * N_ + m]);
}

// point_relation = normalized energy row n==0
__global__ void prcopy_kernel(const float* __restrict__ energy, float* __restrict__ outpr)
{
    const int i = blockIdx.x * blockDim.x + threadIdx.x;
    if (i >= B_ * N_) return;
    const int b = i >> 9, m = i & (N_ - 1);
    outpr[i] = energy[(long)b * N_ * N_ + m];
}

__global__ void sub_kernel(const float* __restrict__ x, const float* __restrict__ xr,
                           float* __restrict__ xd, long n)
{
    const long i = (long)blockIdx.x * blockDim.x + threadIdx.x;
    if (i < n) xd[i] = x[i] - xr[i];
}

// Per-channel batch-norm stats over (B, N).
__global__ __launch_bounds__(256) void bnstats_kernel(
    const float* __restrict__ z, float* __restrict__ mean, float* __restrict__ var)
{
    const int o = blockIdx.x, t = threadIdx.x;
    __shared__ float s1[256], s2[256];
    float a = 0.f, q = 0.f;
    for (int j = t; j < B_ * N_; j += 256) {
        const int bb = j >> 9, m = j & (N_ - 1);
        const float v = z[(long)bb * C_ * N_ + (long)o * N_ + m];
        a += v; q += v * v;
    }
    s1[t] = a; s2[t] = q;
    __syncthreads();
    for (int s = 128; s > 0; s >>= 1) {
        if (t < s) { s1[t] += s1[t + s]; s2[t] += s2[t + s]; }
        __syncthreads();
    }
    if (t == 0) {
        const float mu = s1[0] / (float)(B_ * N_);
        mean[o] = mu;
        var[o]  = s2[0] / (float)(B_ * N_) - mu * mu;
    }
}

// out = x + gelu_exact(bn(z))
__global__ void final_kernel(const float* __restrict__ x, const float* __restrict__ z,
                             const float* __restrict__ mean, const float* __restrict__ var,
                             const float* __restrict__ gamma, const float* __restrict__ beta,
                             float* __restrict__ out)
{
    const long i = (long)blockIdx.x * blockDim.x + threadIdx.x;
    if (i >= (long)B_ * C_ * N_) return;
    const int o = (int)((i / N_) % C_);
    const float y = gamma[o] * (z[i] - mean[o]) * rsqrtf(var[o] + 1e-5f) + beta[o];
    const float g = 0.5f * y * (1.f + erff(y * 0.70710678118654752f));
    out[i] = x[i] + g;
}

// ---------------------------------------------------------------------------
extern "C" void kernel_launch(void* const* d_in, const int* in_sizes, int n_in,
                              void* d_out, int out_size, void* d_ws, size_t ws_size,
                              hipStream_t stream)
{
    const float* x       = (const float*)d_in[0];
    const float* pca     = (const float*)d_in[1];
    const float* a_lrf   = (const float*)d_in[2];
    const float* a_pca   = (const float*)d_in[3];
    const float* Wkp     = (const float*)d_in[4];
    const float* Wk      = (const float*)d_in[5];
    const float* Wv      = (const float*)d_in[6];
    const float* Wvp     = (const float*)d_in[7];
    const float* e1W     = (const float*)d_in[8];
    const float* e1b     = (const float*)d_in[9];
    const float* e2W     = (const float*)d_in[10];
    const float* e2b     = (const float*)d_in[11];
    const float* linW    = (const float*)d_in[12];
    const float* linb    = (const float*)d_in[13];
    const float* gamma   = (const float*)d_in[14];
    const float* beta    = (const float*)d_in[15];
    float* out = (float*)d_out;                    // [B*C*N] out, then [B*N] point_relation

    float* ws = (float*)d_ws;
    float* xq     = ws;                 // B*N*CQ            = 131072
    float* q1     = xq     + 131072;    // B*N*CQ            = 131072
    float* qb     = q1     + 131072;    // B*N               = 2048
    float* ksum   = qb     + 2048;      // B*CQ*N            = 131072
    float* vbar   = ksum   + 131072;    // B*C*N             = 524288
    float* energy = vbar   + 524288;    // B*N*N             = 1048576
    float* colsum = energy + 1048576;   // B*N               = 2048
    float* xr     = colsum + 2048;      // B*C*N             = 524288
    float* xd     = xr     + 524288;    // B*C*N             = 524288
    float* zb     = xd     + 524288;    // B*C*N             = 524288
    float* bnmean = zb     + 524288;    // C                 = 256
    float* bnvar  = bnmean + 256;       // C                 = 256

    const dim3 blk(128);

    // x_q[b,n,o] = sum_c Wkp[o,c] x[b,c,n]     (M=N_ rows n, N=CQ cols o, K=C)
    gemm64_wmma<<<dim3(CQ_/64, N_/64, B_), blk, 0, stream>>>(
        x, 1, N_, (long)C_*N_,  Wkp, 1, C_, 0,
        xq, CQ_, 1, (long)N_*CQ_,  N_, CQ_, C_, 1.f, 0, nullptr);

    // Ksum[b,c,m]  = Wk @ x   ;  += Wkp @ pca  (M=CQ, N=N_, K=C)
    gemm64_wmma<<<dim3(N_/64, CQ_/64, B_), blk, 0, stream>>>(
        Wk, C_, 1, 0,  x, N_, 1, (long)C_*N_,
        ksum, N_, 1, (long)CQ_*N_,  CQ_, N_, C_, 1.f, 0, nullptr);
    gemm64_wmma<<<dim3(N_/64, CQ_/64, B_), blk, 0, stream>>>(
        Wkp, C_, 1, 0,  pca, N_, 1, (long)C_*N_,
        ksum, N_, 1, (long)CQ_*N_,  CQ_, N_, C_, 1.f, 1, nullptr);
    // Ksum += e2^T (angle_pca embedding fold)
    e2_accum_kernel<<<B_*N_, 64, 0, stream>>>(a_pca, e2W, e2b, ksum);

    // q1 = x_q @ emb1_W  (M=N_, N=CQ, K=CQ) ; qb = x_q . emb1_b
    gemm64_wmma<<<dim3(CQ_/64, N_/64, B_), blk, 0, stream>>>(
        xq, CQ_, 1, (long)N_*CQ_,  e1W, CQ_, 1, 0,
        q1, CQ_, 1, (long)N_*CQ_,  N_, CQ_, CQ_, 1.f, 0, nullptr);
    qb_kernel<<<(B_*N_ + 255)/256, 256, 0, stream>>>(xq, e1b, qb);

    // energy = x_q @ Ksum  (M=N_, N=N_, K=CQ)
    gemm64_wmma<<<dim3(N_/64, N_/64, B_), blk, 0, stream>>>(
        xq, CQ_, 1, (long)N_*CQ_,  ksum, N_, 1, (long)CQ_*N_,
        energy, N_, 1, (long)N_*N_,  N_, N_, CQ_, 1.f, 0, nullptr);

    // += folded angle_lrf term; softmax; column normalize
    angle_add_kernel<<<B_*N_, 256, 0, stream>>>(a_lrf, q1, qb, energy);
    softmax_kernel<<<B_*N_, 256, 0, stream>>>(energy);
    colsum_kernel<<<(B_*N_ + 255)/256, 256, 0, stream>>>(energy, colsum);
    colnorm_kernel<<<(B_*N_*N_ + 255)/256, 256, 0, stream>>>(energy, colsum);
    prcopy_kernel<<<(B_*N_ + 255)/256, 256, 0, stream>>>(energy, out + (long)B_*C_*N_);

    // vbar = 0.5*Wv@x + 0.5*Wvp@pca  (M=C, N=N_, K=C)
    gemm64_wmma<<<dim3(N_/64, C_/64, B_), blk, 0, stream>>>(
        Wv, C_, 1, 0,  x, N_, 1, (long)C_*N_,
        vbar, N_, 1, (long)C_*N_,  C_, N_, C_, 0.5f, 0, nullptr);
    gemm64_wmma<<<dim3(N_/64, C_/64, B_), blk, 0, stream>>>(
        Wvp, C_, 1, 0,  pca, N_, 1, (long)C_*N_,
        vbar, N_, 1, (long)C_*N_,  C_, N_, C_, 0.5f, 1, nullptr);

    // x_r = vbar @ En  (M=C, N=N_, K=N_)
    gemm64_wmma<<<dim3(N_/64, C_/64, B_), blk, 0, stream>>>(
        vbar, N_, 1, (long)C_*N_,  energy, N_, 1, (long)N_*N_,
        xr, N_, 1, (long)C_*N_,  C_, N_, N_, 1.f, 0, nullptr);

    // xd = x - x_r ; z = lin_W @ xd + lin_b
    sub_kernel<<<((long)B_*C_*N_ + 255)/256, 256, 0, stream>>>(x, xr, xd, (long)B_*C_*N_);
    gemm64_wmma<<<dim3(N_/64, C_/64, B_), blk, 0, stream>>>(
        linW, C_, 1, 0,  xd, N_, 1, (long)C_*N_,
        zb, N_, 1, (long)C_*N_,  C_, N_, C_, 1.f, 0, linb);

    // batch-norm over (B, N) per channel, then out = x + gelu(y)
    bnstats_kernel<<<C_, 256, 0, stream>>>(zb, bnmean, bnvar);
    final_kernel<<<((long)B_*C_*N_ + 255)/256, 256, 0, stream>>>(
        x, zb, bnmean, bnvar, gamma, beta, out);
}